// MultiSeqQueryGenerator_4922032521973
// MI455X (gfx1250) — compile-verified
//
#include <hip/hip_runtime.h>
#include <hip/hip_bf16.h>

typedef __attribute__((ext_vector_type(2))) float v2f;
typedef __attribute__((ext_vector_type(8))) float v8f;

#define S_ 4
#define B_ 256
#define L_ 256
#define D_ 512
#define NQ_ 4
#define H_ 2048
#define G_ 3072
#define EPS_ 1e-5f

// ---------------------------------------------------------------------------
// Kernel 1: masked mean pooling over L.  grid = S*B blocks, 256 threads.
// Each thread owns columns d = t and d = t + 256.
// ---------------------------------------------------------------------------
__global__ __launch_bounds__(256) void pool_kernel(
    const float* __restrict__ seq,          // (S,B,L,D)
    const unsigned char* __restrict__ mask, // (S,B,L) bool
    float* __restrict__ pooled)             // (S,B,D)
{
    const int blk = blockIdx.x;             // s*B + b
    const int t = threadIdx.x;
    const size_t base = (size_t)blk * L_ * D_;
    const size_t mbase = (size_t)blk * L_;
    float a0 = 0.f, a1 = 0.f;
    int cnt = 0;
    for (int l = 0; l < L_; ++l) {
        const unsigned char pad = mask[mbase + l];
        if (!pad) {
            const size_t row = base + (size_t)l * D_;
            a0 += seq[row + t];
            a1 += seq[row + 256 + t];
            ++cnt;
        }
    }
    const float denom = 1.f / (float)(cnt > 0 ? cnt : 1);
    pooled[(size_t)blk * D_ + t]       = a0 * denom;
    pooled[(size_t)blk * D_ + 256 + t] = a1 * denom;
}

// ---------------------------------------------------------------------------
// Block-wide sum over 256 threads via LDS tree.
// ---------------------------------------------------------------------------
__device__ inline float block_sum(float v, float* red) {
    const int t = threadIdx.x;
    red[t] = v;
    __syncthreads();
    #pragma unroll
    for (int off = 128; off > 0; off >>= 1) {
        if (t < off) red[t] += red[t + off];
        __syncthreads();
    }
    const float r = red[0];
    __syncthreads();
    return r;
}

// ---------------------------------------------------------------------------
// Kernel 2: stat matvec + stat LN + gi assembly + G-wide LN.
// grid = S*B blocks, 256 threads.  Thread t owns stat dims {t, t+256} and
// gi elements {i*256 + t : i<12}.
// ---------------------------------------------------------------------------
__global__ __launch_bounds__(256) void gi_kernel(
    const float* __restrict__ ns,       // (B, M*D) = (256, 2048)
    const float* __restrict__ pooled,   // (S,B,D)
    const float* __restrict__ stats,    // (S,B,6)
    const float* __restrict__ statW,    // (S,6,D)
    const float* __restrict__ statb,    // (S,D)
    const float* __restrict__ slng,     // (S,D)
    const float* __restrict__ slnb,     // (S,D)
    const float* __restrict__ ging,     // (G)
    const float* __restrict__ ginb,     // (G)
    float* __restrict__ giw)            // (S,B,G)
{
    __shared__ float red[256];
    const int blk = blockIdx.x;
    const int s = blk / B_;
    const int b = blk % B_;
    const int t = threadIdx.x;

    float sf[6];
    #pragma unroll
    for (int f = 0; f < 6; ++f) sf[f] = stats[(size_t)blk * 6 + f];

    const int d0 = t, d1 = t + 256;
    float r0 = statb[s * D_ + d0];
    float r1 = statb[s * D_ + d1];
    #pragma unroll
    for (int f = 0; f < 6; ++f) {
        const float* wrow = statW + ((size_t)s * 6 + f) * D_;
        r0 += sf[f] * wrow[d0];
        r1 += sf[f] * wrow[d1];
    }
    // LN over D for stat
    const float s1 = block_sum(r0 + r1, red);
    const float s2 = block_sum(r0 * r0 + r1 * r1, red);
    const float mu = s1 * (1.f / (float)D_);
    const float var = s2 * (1.f / (float)D_) - mu * mu;
    const float rst = rsqrtf(var + EPS_);
    const float n0 = (r0 - mu) * rst * slng[s * D_ + d0] + slnb[s * D_ + d0];
    const float n1 = (r1 - mu) * rst * slng[s * D_ + d1] + slnb[s * D_ + d1];

    // assemble gi values (12 per thread)
    float vals[12];
    #pragma unroll
    for (int i = 0; i < 8; ++i)
        vals[i] = ns[(size_t)b * 2048 + i * 256 + t];
    vals[8]  = pooled[(size_t)blk * D_ + t];
    vals[9]  = pooled[(size_t)blk * D_ + 256 + t];
    vals[10] = n0;
    vals[11] = n1;

    float ls = 0.f, ls2 = 0.f;
    #pragma unroll
    for (int i = 0; i < 12; ++i) { ls += vals[i]; ls2 += vals[i] * vals[i]; }
    const float gs1 = block_sum(ls, red);
    const float gs2 = block_sum(ls2, red);
    const float gmu = gs1 * (1.f / (float)G_);
    const float gvar = gs2 * (1.f / (float)G_) - gmu * gmu;
    const float grst = rsqrtf(gvar + EPS_);

    float* orow = giw + (size_t)blk * G_;
    #pragma unroll
    for (int i = 0; i < 12; ++i) {
        const int j = i * 256 + t;
        orow[j] = (vals[i] - gmu) * grst * ging[j] + ginb[j];
    }
}

// ---------------------------------------------------------------------------
// Kernel 3: GEMM1 + bias + SiLU.  h[sq] = silu(gi[s] @ W1[sq] + b1[sq]).
// grid = (H/128, B/128, S*Nq); 256 threads = 8 waves.
// Workgroup tile: 128(B) x 128(H).  Wave tile: 32 x 64 = 2x4 WMMA tiles.
// ---------------------------------------------------------------------------
__global__ __launch_bounds__(256) void gemm1_kernel(
    const float* __restrict__ gi,   // (S,B,G)
    const float* __restrict__ W1,   // (S,Nq,G,H)
    const float* __restrict__ b1,   // (S,Nq,H)
    float* __restrict__ h)          // (S*Nq, B, H)
{
    __shared__ float As[128][18];   // A tile, [row][k]
    __shared__ float Bs[128][18];   // B tile transposed, [n][k]

    const int t = threadIdx.x;
    const int sq = blockIdx.z;
    const int s = sq >> 2;
    const int n0g = blockIdx.x * 128;
    const int m0g = blockIdx.y * 128;
    const int wid = t >> 5;
    const int lane = t & 31;
    const int half = lane >> 4;
    const int lm = lane & 15;
    const int mw = (wid >> 1) * 32;  // 0,32,64,96
    const int nw = (wid & 1) * 64;   // 0,64

    v8f acc[2][4];
    #pragma unroll
    for (int i = 0; i < 2; ++i)
        #pragma unroll
        for (int j = 0; j < 4; ++j)
            acc[i][j] = {};

    const float* AG = gi + ((size_t)s * B_ + m0g) * G_;
    const float* BG = W1 + (size_t)sq * G_ * H_;

    for (int kt = 0; kt < G_; kt += 16) {
        // stage A tile 128x16 (coalesced: consecutive threads -> consecutive k)
        #pragma unroll
        for (int i = 0; i < 8; ++i) {
            const int e = i * 256 + t;
            const int r = e >> 4, c = e & 15;
            As[r][c] = AG[(size_t)r * G_ + kt + c];
        }
        // stage B tile 16x128, transposed into LDS (consecutive threads -> consecutive n)
        #pragma unroll
        for (int i = 0; i < 8; ++i) {
            const int e = i * 256 + t;
            const int kk = e >> 7, n = e & 127;
            Bs[n][kk] = BG[(size_t)(kt + kk) * H_ + n0g + n];
        }
        if (kt + 16 < G_)
            __builtin_prefetch(&BG[(size_t)(kt + 16) * H_ + n0g + (t & 127)], 0, 0);
        __syncthreads();

        #pragma unroll
        for (int kc = 0; kc < 4; ++kc) {
            const int kb = kc * 4 + 2 * half;
            v2f a[2], b[4];
            #pragma unroll
            for (int mt = 0; mt < 2; ++mt) {
                const float2 p = *(const float2*)&As[mw + mt * 16 + lm][kb];
                a[mt].x = p.x; a[mt].y = p.y;
            }
            #pragma unroll
            for (int nt = 0; nt < 4; ++nt) {
                const float2 p = *(const float2*)&Bs[nw + nt * 16 + lm][kb];
                b[nt].x = p.x; b[nt].y = p.y;
            }
            #pragma unroll
            for (int mt = 0; mt < 2; ++mt)
                #pragma unroll
                for (int nt = 0; nt < 4; ++nt)
                    acc[mt][nt] = __builtin_amdgcn_wmma_f32_16x16x4_f32(
                        false, a[mt], false, b[nt], (short)0, acc[mt][nt],
                        false, false);
        }
        __syncthreads();
    }

    // epilogue: + b1, SiLU, store h
    #pragma unroll
    for (int mt = 0; mt < 2; ++mt) {
        #pragma unroll
        for (int nt = 0; nt < 4; ++nt) {
            const int n = n0g + nw + nt * 16 + lm;
            const float bias = b1[(size_t)sq * H_ + n];
            #pragma unroll
            for (int v = 0; v < 8; ++v) {
                const int m = m0g + mw + mt * 16 + 8 * half + v;
                float x = acc[mt][nt][v] + bias;
                x = x / (1.f + __expf(-x));
                h[((size_t)sq * B_ + m) * H_ + n] = x;
            }
        }
    }
}

// ---------------------------------------------------------------------------
// Kernel 4: GEMM2 + bias + LN + transpose-out.
// grid = (B/16, S*Nq); 256 threads = 8 waves; workgroup = 16 full rows (D=512)
// so row LN can be reduced in LDS.  Wave owns a 64-col slice (4 WMMA tiles).
// ---------------------------------------------------------------------------
__global__ __launch_bounds__(256) void gemm2_kernel(
    const float* __restrict__ h,    // (S*Nq, B, H)
    const float* __restrict__ W2,   // (S,Nq,H,D)
    const float* __restrict__ b2,   // (S,Nq,D)
    const float* __restrict__ lng,  // (S,Nq,D)
    const float* __restrict__ lnb,  // (S,Nq,D)
    float* __restrict__ out)        // (S,B,Nq,D)
{
    __shared__ float As[16][20];
    __shared__ float lsum[16], lsq[16], lmean[16], lrstd[16];

    const int t = threadIdx.x;
    const int sq = blockIdx.y;
    const int s = sq >> 2;
    const int q = sq & 3;
    const int m0 = blockIdx.x * 16;
    const int wid = t >> 5;
    const int lane = t & 31;
    const int half = lane >> 4;
    const int lm = lane & 15;
    const int nwv = wid * 64;

    v8f acc[4];
    #pragma unroll
    for (int j = 0; j < 4; ++j) acc[j] = {};

    const float* AG = h + ((size_t)sq * B_ + m0) * H_;
    const float* BG = W2 + (size_t)sq * H_ * D_;

    for (int kt = 0; kt < H_; kt += 16) {
        // stage A tile 16x16
        {
            const int r = t >> 4, c = t & 15;
            As[r][c] = AG[(size_t)r * H_ + kt + c];
        }
        __syncthreads();
        #pragma unroll
        for (int kc = 0; kc < 4; ++kc) {
            const int kb = kc * 4 + 2 * half;
            const float2 p = *(const float2*)&As[lm][kb];
            v2f a; a.x = p.x; a.y = p.y;
            #pragma unroll
            for (int nt = 0; nt < 4; ++nt) {
                const int n = nwv + nt * 16 + lm;
                const float* bp = BG + (size_t)(kt + kb) * D_ + n;
                v2f b; b.x = bp[0]; b.y = bp[D_];
                acc[nt] = __builtin_amdgcn_wmma_f32_16x16x4_f32(
                    false, a, false, b, (short)0, acc[nt], false, false);
            }
        }
        __syncthreads();
    }

    // epilogue: + b2, row LayerNorm over D=512, write (S,B,Nq,D)
    if (t < 16) { lsum[t] = 0.f; lsq[t] = 0.f; }
    __syncthreads();

    float bias[4];
    #pragma unroll
    for (int nt = 0; nt < 4; ++nt)
        bias[nt] = b2[(size_t)sq * D_ + nwv + nt * 16 + lm];

    #pragma unroll
    for (int v = 0; v < 8; ++v) {
        const int m = v + 8 * half;
        float ps = 0.f, ps2 = 0.f;
        #pragma unroll
        for (int nt = 0; nt < 4; ++nt) {
            const float x = acc[nt][v] + bias[nt];
            ps += x; ps2 += x * x;
        }
        atomicAdd(&lsum[m], ps);
        atomicAdd(&lsq[m], ps2);
    }
    __syncthreads();
    if (t < 16) {
        const float mu = lsum[t] * (1.f / (float)D_);
        const float var = lsq[t] * (1.f / (float)D_) - mu * mu;
        lmean[t] = mu;
        lrstd[t] = rsqrtf(var + EPS_);
    }
    __syncthreads();

    #pragma unroll
    for (int nt = 0; nt < 4; ++nt) {
        const int n = nwv + nt * 16 + lm;
        const float g = lng[(size_t)sq * D_ + n];
        const float be = lnb[(size_t)sq * D_ + n];
        #pragma unroll
        for (int v = 0; v < 8; ++v) {
            const int m = v + 8 * half;
            const float x = acc[nt][v] + bias[nt];
            const float y = (x - lmean[m]) * lrstd[m] * g + be;
            const int brow = m0 + m;
            out[(((size_t)s * B_ + brow) * NQ_ + q) * D_ + n] = y;
        }
    }
}

// ---------------------------------------------------------------------------
extern "C" void kernel_launch(void* const* d_in, const int* in_sizes, int n_in,
                              void* d_out, int out_size, void* d_ws, size_t ws_size,
                              hipStream_t stream) {
    const float* ns      = (const float*)d_in[0];
    const float* seq     = (const float*)d_in[1];
    const unsigned char* mask = (const unsigned char*)d_in[2];
    const float* stats   = (const float*)d_in[3];
    const float* ging    = (const float*)d_in[4];
    const float* ginb    = (const float*)d_in[5];
    const float* statW   = (const float*)d_in[6];
    const float* statb   = (const float*)d_in[7];
    const float* slng    = (const float*)d_in[8];
    const float* slnb    = (const float*)d_in[9];
    const float* W1      = (const float*)d_in[10];
    const float* b1      = (const float*)d_in[11];
    const float* W2      = (const float*)d_in[12];
    const float* b2      = (const float*)d_in[13];
    const float* lng     = (const float*)d_in[14];
    const float* lnb     = (const float*)d_in[15];
    float* out = (float*)d_out;

    float* ws = (float*)d_ws;
    float* pooled = ws;                                   // S*B*D   =  524288 f
    float* giw    = pooled + (size_t)S_ * B_ * D_;        // S*B*G   = 3145728 f
    float* hw     = giw + (size_t)S_ * B_ * G_;           // SQ*B*H  = 8388608 f

    pool_kernel<<<S_ * B_, 256, 0, stream>>>(seq, mask, pooled);
    gi_kernel<<<S_ * B_, 256, 0, stream>>>(ns, pooled, stats, statW, statb,
                                           slng, slnb, ging, ginb, giw);
    dim3 g1(H_ / 128, B_ / 128, S_ * NQ_);   // (16, 2, 16)
    gemm1_kernel<<<g1, 256, 0, stream>>>(giw, W1, b1, hw);
    dim3 g2(B_ / 16, S_ * NQ_);              // (16, 16)
    gemm2_kernel<<<g2, 256, 0, stream>>>(hw, W2, b2, lng, lnb, out);
}